// GNNWithAttention_37967510896823
// MI455X (gfx1250) — compile-verified
//
#include <hip/hip_runtime.h>

typedef float v2f __attribute__((ext_vector_type(2)));
typedef float v8f __attribute__((ext_vector_type(8)));

#define NEG_SLOPE 0.2f

// ---------------------------------------------------------------------------
// FP32 WMMA GEMM: C[M x Nc] = A[M x K] * Bt^T, Bt is B transposed [Nc x K].
// Block = 8 waves = 8 consecutive M-tiles sharing one 64-column N-group.
// The 64-column B panel (contiguous in Bt) is staged in LDS in KC=128 chunks;
// inner loop: 1 global b64 (A, double-buffered) + 4 ds_load_b64 (B).
// LDS stride KC+4 -> ds accesses tile all 64 banks exactly once (no conflict).
// Requires M%16==0, K%128==0 or K%KC chunking exact, Nc%64==0.
// VGPR layouts per CDNA5 ISA 7.12.2:
//   A 16x4 f32 : lane m = lane&15 ; v0 = A[m][k0+2*(lane>>4)], v1 = +1
//   B 4x16 f32 : lane n = lane&15 ; v0 = B[k0+2*(lane>>4)][n], v1 = +1
//   C/D 16x16  : lane n = lane&15 ; vgpr v -> row m = v + 8*(lane>>4)
// ---------------------------------------------------------------------------
__global__ void gat_gemm_wmma_f32(const float* __restrict__ A,
                                  const float* __restrict__ Bt,
                                  float* __restrict__ C,
                                  int M, int K, int Nc) {
  const int KC  = 128;          // K staged per chunk
  const int STR = KC + 4;       // LDS column stride (bank-conflict-free)
  __shared__ float ldsB[64 * STR];          // 33 KB

  int groupsN   = Nc >> 6;                  // Nc / 64
  int mChunk    = blockIdx.x / groupsN;
  int nGroup    = blockIdx.x % groupsN;
  int waveInBlk = threadIdx.x >> 5;
  int lane      = threadIdx.x & 31;
  int tileM     = mChunk * 8 + waveInBlk;
  bool valid    = (tileM * 16 < M);         // wave-uniform

  int half = lane >> 4;                     // 0 | 1
  int r    = lane & 15;                     // A-row / B-col / C-col

  v8f acc0 = {0,0,0,0,0,0,0,0};
  v8f acc1 = acc0, acc2 = acc0, acc3 = acc0;

  const float* arow = A + (size_t)((valid ? tileM : 0) * 16 + r) * K + 2 * half;
  const float* bsrc = Bt + (size_t)nGroup * 64 * K;   // 64 contiguous columns
  const float* lb   = ldsB + (size_t)r * STR + 2 * half;

  for (int kc = 0; kc < K; kc += KC) {
    __syncthreads();                        // previous chunk fully consumed
    // ---- stage 64 x KC B-panel into LDS (float4, coalesced) ----
    for (int i = threadIdx.x; i < 64 * (KC / 4); i += 256) {
      int c = i / (KC / 4), kq = i % (KC / 4);
      ((float4*)(ldsB + c * STR))[kq] =
          ((const float4*)(bsrc + (size_t)c * K + kc))[kq];
    }
    __syncthreads();

    if (valid) {
#define DO_WMMA(av, kk)                                                        \
      {                                                                        \
        v2f b0, b1, b2, b3;                                                    \
        b0.x = lb[kk];            b0.y = lb[(kk) + 1];                         \
        b1.x = lb[16 * STR + kk]; b1.y = lb[16 * STR + (kk) + 1];              \
        b2.x = lb[32 * STR + kk]; b2.y = lb[32 * STR + (kk) + 1];              \
        b3.x = lb[48 * STR + kk]; b3.y = lb[48 * STR + (kk) + 1];              \
        acc0 = __builtin_amdgcn_wmma_f32_16x16x4_f32(false, av, false, b0, (short)0, acc0, false, false); \
        acc1 = __builtin_amdgcn_wmma_f32_16x16x4_f32(false, av, false, b1, (short)0, acc1, false, false); \
        acc2 = __builtin_amdgcn_wmma_f32_16x16x4_f32(false, av, false, b2, (short)0, acc2, false, false); \
        acc3 = __builtin_amdgcn_wmma_f32_16x16x4_f32(false, av, false, b3, (short)0, acc3, false, false); \
      }
      v2f a0; a0.x = arow[kc]; a0.y = arow[kc + 1];      // current A frag
      #pragma unroll 4
      for (int k0 = 0; k0 < KC - 4; k0 += 4) {
        v2f a1;                                          // prefetch next A
        a1.x = arow[kc + k0 + 4]; a1.y = arow[kc + k0 + 5];
        DO_WMMA(a0, k0)
        a0 = a1;
      }
      DO_WMMA(a0, KC - 4)                                // drain
#undef DO_WMMA
    }
  }

  if (valid) {
    int mbase = tileM * 16 + 8 * half;
    float* c0 = C + (size_t)mbase * Nc + (size_t)nGroup * 64 + r;
    #pragma unroll
    for (int v = 0; v < 8; ++v) {
      size_t row = (size_t)v * Nc;
      c0[row]      = acc0[v];
      c0[row + 16] = acc1[v];
      c0[row + 32] = acc2[v];
      c0[row + 48] = acc3[v];
    }
  }
}

// Wt[n*K + k] = W[k*Nc + n]  (writes coalesced; tiny matrices, L2-resident)
__global__ void gat_transpose(const float* __restrict__ W, float* __restrict__ Wt,
                              int K, int Nc) {
  int idx = blockIdx.x * blockDim.x + threadIdx.x;
  if (idx >= K * Nc) return;
  int n = idx / K, k = idx % K;
  Wt[idx] = W[(size_t)k * Nc + n];
}

// el/er per (node, head): wave-per-(n,h), coalesced b64 loads + shfl reduce
__global__ void gat_dots(const float* __restrict__ feat,
                         const float* __restrict__ al,
                         const float* __restrict__ ar,
                         float* __restrict__ el, float* __restrict__ er,
                         int Nn, int H, int D) {
  int wave = (int)((blockIdx.x * blockDim.x + threadIdx.x) >> 5);
  int lane = threadIdx.x & 31;
  if (wave >= Nn * H) return;
  int n = wave / H, h = wave % H;
  const float* f = feat + (size_t)n * H * D + (size_t)h * D;
  const float* l = al + (size_t)h * D;
  const float* r = ar + (size_t)h * D;
  float sl = 0.f, sr = 0.f;
  for (int d = lane * 2; d < D; d += 64) {      // D=64 -> one pair per lane
    float v0 = f[d], v1 = f[d + 1];
    sl += v0 * l[d] + v1 * l[d + 1];
    sr += v0 * r[d] + v1 * r[d + 1];
  }
  #pragma unroll
  for (int off = 16; off; off >>= 1) {
    sl += __shfl_xor(sl, off);
    sr += __shfl_xor(sr, off);
  }
  if (lane == 0) { el[wave] = sl; er[wave] = sr; }
}

// init segment-max keys / denominators to "empty"; rst to broadcast bias
__global__ void gat_init(unsigned* __restrict__ emax, float* __restrict__ denom,
                         float* __restrict__ rst, const float* __restrict__ bias,
                         int NH, int NHD, int HD) {
  int idx = blockIdx.x * blockDim.x + threadIdx.x;
  if (idx < NH) { emax[idx] = 0u; denom[idx] = 0.f; }
  if (idx < NHD) rst[idx] = bias[idx % HD];
}

// monotone float <-> uint key (ascending order preserved; 0u < key(-inf))
__device__ __forceinline__ unsigned fkey(float f) {
  int i = __float_as_int(f);
  unsigned u = (unsigned)i;
  return (i >= 0) ? (u | 0x80000000u) : ~u;
}
__device__ __forceinline__ float fkeyinv(unsigned u) {
  int i = (u & 0x80000000u) ? (int)(u & 0x7FFFFFFFu) : (int)~u;
  return __int_as_float(i);
}

// per (edge, head): e = leaky_relu(el[src]+er[dst]); segment max via atomicMax
__global__ void gat_edge_logits(const int* __restrict__ src, const int* __restrict__ dst,
                                const float* __restrict__ el, const float* __restrict__ er,
                                float* __restrict__ ebuf, unsigned* __restrict__ emax,
                                int Ne, int H) {
  int idx = blockIdx.x * blockDim.x + threadIdx.x;
  if (idx >= Ne * H) return;
  int e = idx / H, h = idx % H;
  int s = src[e], d = dst[e];
  float x = el[s * H + h] + er[d * H + h];
  x = (x > 0.f) ? x : NEG_SLOPE * x;
  ebuf[idx] = x;
  atomicMax(&emax[d * H + h], fkey(x));
}

// per (edge, head): ex = exp(e - max[dst]); denom[dst] += ex  (ebuf overwritten)
__global__ void gat_edge_exp(const int* __restrict__ dst,
                             float* __restrict__ ebuf,
                             const unsigned* __restrict__ emax,
                             float* __restrict__ denom, int Ne, int H) {
  int idx = blockIdx.x * blockDim.x + threadIdx.x;
  if (idx >= Ne * H) return;
  int e = idx / H, h = idx % H;
  int d = dst[e];
  float m = fkeyinv(emax[d * H + h]);
  float ex = expf(ebuf[idx] - m);
  ebuf[idx] = ex;
  atomicAdd(&denom[d * H + h], ex);
}

// per (edge, h*D+j): rst[dst] += (ex/denom[dst]) * feat[src]   (coalesced per edge)
__global__ void gat_edge_aggr(const int* __restrict__ src, const int* __restrict__ dst,
                              const float* __restrict__ ex, const float* __restrict__ denom,
                              const float* __restrict__ feat, float* __restrict__ rst,
                              int Ne, int H, int D) {
  int HD = H * D;
  long long gid = (long long)blockIdx.x * blockDim.x + threadIdx.x;
  if (gid >= (long long)Ne * HD) return;
  int e = (int)(gid / HD);
  int j = (int)(gid % HD);
  int h = j / D;
  int s = src[e], d = dst[e];
  float alpha = ex[(size_t)e * H + h] / denom[d * H + h];
  atomicAdd(&rst[(size_t)d * HD + j], alpha * feat[(size_t)s * HD + j]);
}

extern "C" void kernel_launch(void* const* d_in, const int* in_sizes, int n_in,
                              void* d_out, int out_size, void* d_ws, size_t ws_size,
                              hipStream_t stream) {
  const int N = 50000, E = 400000;
  const float* features = (const float*)d_in[0];
  const int*   src      = (const int*)d_in[1];
  const int*   dst      = (const int*)d_in[2];
  const float* W1       = (const float*)d_in[3];
  const float* al1      = (const float*)d_in[4];
  const float* ar1      = (const float*)d_in[5];
  const float* b1       = (const float*)d_in[6];
  const float* W2       = (const float*)d_in[7];
  const float* al2      = (const float*)d_in[8];
  const float* ar2      = (const float*)d_in[9];
  const float* b2       = (const float*)d_in[10];
  float* out = (float*)d_out;

  float* ws = (float*)d_ws;
  size_t o = 0;
  float* feat1 = ws + o; o += (size_t)N * 256;   // layer1 projected feats
  float* h1    = ws + o; o += (size_t)N * 256;   // layer1 output (layer2 input)
  float* feat2 = ws + o; o += (size_t)N * 64;    // layer2 projected feats
  float* el    = ws + o; o += (size_t)N * 4;
  float* er    = ws + o; o += (size_t)N * 4;
  unsigned* emax = (unsigned*)(ws + o); o += (size_t)N * 4;
  float* denom = ws + o; o += (size_t)N * 4;
  float* ebuf  = ws + o; o += (size_t)E * 4;     // logits, then exp(logits)
  float* W1t   = ws + o; o += (size_t)128 * 256; // W1 transposed [256 x 128]
  float* W2t   = ws + o; o += (size_t)256 * 64;  // W2 transposed [64 x 256]

  const dim3 blk(256);
  gat_transpose<<<(128 * 256 + 255) / 256, blk, 0, stream>>>(W1, W1t, 128, 256);
  gat_transpose<<<(256 * 64 + 255) / 256, blk, 0, stream>>>(W2, W2t, 256, 64);

  int mChunks = (N / 16 + 7) / 8;                // 8 M-tiles per block

  // ---------------- Layer 1 (H=4, D=64, HD=256) ----------------
  {
    gat_gemm_wmma_f32<<<mChunks * (256 / 64), blk, 0, stream>>>(features, W1t, feat1, N, 128, 256);
    gat_dots<<<(N * 4 * 32 + 255) / 256, blk, 0, stream>>>(feat1, al1, ar1, el, er, N, 4, 64);
    gat_init<<<((size_t)N * 256 + 255) / 256, blk, 0, stream>>>(emax, denom, h1, b1, N * 4, N * 256, 256);
    gat_edge_logits<<<(E * 4 + 255) / 256, blk, 0, stream>>>(src, dst, el, er, ebuf, emax, E, 4);
    gat_edge_exp<<<(E * 4 + 255) / 256, blk, 0, stream>>>(dst, ebuf, emax, denom, E, 4);
    gat_edge_aggr<<<((size_t)E * 256 + 255) / 256, blk, 0, stream>>>(src, dst, ebuf, denom, feat1, h1, E, 4, 64);
  }
  // ---------------- Layer 2 (H=1, D=64, HD=64) ----------------
  {
    gat_gemm_wmma_f32<<<mChunks * (64 / 64), blk, 0, stream>>>(h1, W2t, feat2, N, 256, 64);
    gat_dots<<<(N * 32 + 255) / 256, blk, 0, stream>>>(feat2, al2, ar2, el, er, N, 1, 64);
    gat_init<<<((size_t)N * 64 + 255) / 256, blk, 0, stream>>>(emax, denom, out, b2, N, N * 64, 64);
    gat_edge_logits<<<(E + 255) / 256, blk, 0, stream>>>(src, dst, el, er, ebuf, emax, E, 1);
    gat_edge_exp<<<(E + 255) / 256, blk, 0, stream>>>(dst, ebuf, emax, denom, E, 1);
    gat_edge_aggr<<<((size_t)E * 64 + 255) / 256, blk, 0, stream>>>(src, dst, ebuf, denom, feat2, out, E, 1, 64);
  }
}